// PointNetEncoder_83451214561579
// MI455X (gfx1250) — compile-verified
//
#include <hip/hip_runtime.h>
#include <hip/hip_bf16.h>

// ---------------- problem constants ----------------
constexpr int   B_  = 4;
constexpr int   N_  = 8192;
constexpr int   K_  = 20;
constexpr float EPS_    = 1e-6f;
constexpr float BNEPS_  = 1e-5f;
constexpr int   NC_ = 3 * N_;          // columns per batch for channel GEMMs (v,n)
constexpr int   PCP_ = 96;             // padded channel stride of GEMM outputs (P/D)

// ---------------- WMMA vector types ----------------
typedef __attribute__((ext_vector_type(2))) float v2f;
typedef __attribute__((ext_vector_type(8))) float v8f;

// ---------------- small helpers ----------------
__device__ __forceinline__ unsigned sortable(float f) {
  unsigned u = __float_as_uint(f);
  return (u & 0x80000000u) ? ~u : (u | 0x80000000u);
}
__device__ __forceinline__ float unsortable(unsigned u) {
  return __uint_as_float((u & 0x80000000u) ? (u & 0x7FFFFFFFu) : ~u);
}

__global__ void fill_f32(float* p, int n, float v) {
  for (int t = blockIdx.x * blockDim.x + threadIdx.x; t < n; t += gridDim.x * blockDim.x) p[t] = v;
}
__global__ void fill_u32(unsigned* p, int n, unsigned v) {
  for (int t = blockIdx.x * blockDim.x + threadIdx.x; t < n; t += gridDim.x * blockDim.x) p[t] = v;
}

// zero the channel-pad rows [C, CP) of a padded activation buffer
__global__ void zero_pad(float* __restrict__ X, int C, int CP) {
  const int per = (CP - C) * 3 * N_;
  const int t = blockIdx.x * blockDim.x + threadIdx.x;
  if (t >= B_ * per) return;
  const int b = t / per, rr = t % per;
  X[(size_t)b * CP * 3 * N_ + (size_t)C * 3 * N_ + rr] = 0.f;
}

// pack W (M x Kd, row stride ldw) into zero-padded Mp16 x Kp4
__global__ void pack_w(const float* __restrict__ W, int ldw, int M, int Kd,
                       float* __restrict__ Wp, int Mp16, int Kp4) {
  const int t = blockIdx.x * blockDim.x + threadIdx.x;
  if (t >= Mp16 * Kp4) return;
  const int r = t / Kp4, k = t % Kp4;
  Wp[t] = (r < M && k < Kd) ? W[r * ldw + k] : 0.f;
}

// ============================================================
// KNN: per-query top-20 (smallest squared distance, self incl.)
// LDS-staged candidate tiles, register insertion list.
// ============================================================
__global__ void __launch_bounds__(256)
knn_kernel(const float* __restrict__ pts, int* __restrict__ idxbuf) {
  __shared__ float sx[256], sy[256], sz[256];
  const int tid = threadIdx.x;
  const int bpb = N_ / 256;
  const int b = blockIdx.x / bpb;
  const int n = (blockIdx.x % bpb) * 256 + tid;
  const size_t qoff = ((size_t)b * N_ + n) * 3;
  const float qx = pts[qoff], qy = pts[qoff + 1], qz = pts[qoff + 2];

  float kd[K_]; int ki[K_];
#pragma unroll
  for (int j = 0; j < K_; ++j) { kd[j] = 3.4e38f; ki[j] = 0; }

  for (int ct = 0; ct < bpb; ++ct) {
    const size_t coff = ((size_t)b * N_ + ct * 256 + tid) * 3;
    __syncthreads();
    sx[tid] = pts[coff]; sy[tid] = pts[coff + 1]; sz[tid] = pts[coff + 2];
    __syncthreads();
    if (ct + 1 < bpb)  // CDNA5 global_prefetch_b8 of next candidate tile
      __builtin_prefetch(&pts[((size_t)b * N_ + (ct + 1) * 256 + tid) * 3], 0, 0);
    for (int j = 0; j < 256; ++j) {
      const float dx = qx - sx[j], dy = qy - sy[j], dz = qz - sz[j];
      const float d2 = dx * dx + dy * dy + dz * dz;
      if (d2 < kd[K_ - 1]) {          // rare path: expected ~k*ln(N/k) per lane
        const int ci = ct * 256 + j;
        bool done = false;
#pragma unroll
        for (int s = K_ - 1; s >= 1; --s) {
          if (!done) {
            if (d2 < kd[s - 1]) { kd[s] = kd[s - 1]; ki[s] = ki[s - 1]; }
            else { kd[s] = d2; ki[s] = ci; done = true; }
          }
        }
        if (!done) { kd[0] = d2; ki[0] = ci; }
      }
    }
  }
#pragma unroll
  for (int j = 0; j < K_; ++j) idxbuf[((size_t)b * N_ + n) * K_ + j] = ki[j];
}

// ============================================================
// wpos layer (fused, two-pass): graph feature (e, ctr, cross),
// 21x3 linear f/d, global BN over ||p||, LLR, mean over K.
// ============================================================
__device__ __forceinline__ void graph_feat(const float* __restrict__ pts,
                                           const int* __restrict__ idxbuf,
                                           int b, int n, int k,
                                           float cx, float cy, float cz,
                                           float& ex, float& ey, float& ez,
                                           float& rx, float& ry, float& rz) {
  const int j = idxbuf[((size_t)b * N_ + n) * K_ + k];
  const size_t o = ((size_t)b * N_ + j) * 3;
  const float nx = pts[o], ny = pts[o + 1], nz = pts[o + 2];
  ex = nx - cx; ey = ny - cy; ez = nz - cz;
  rx = ny * cz - nz * cy; ry = nz * cx - nx * cz; rz = nx * cy - ny * cx;  // cross(nbr, ctr)
}

__global__ void __launch_bounds__(256)
wpos_stats(const float* __restrict__ pts, const int* __restrict__ idxbuf,
           const float* __restrict__ wf, float* __restrict__ stats) {
  __shared__ float ssum[42];
  const int tid = threadIdx.x;
  if (tid < 42) ssum[tid] = 0.f;
  __syncthreads();
  const int bpb = N_ / 256;
  const int b = blockIdx.x / bpb;
  const int n = (blockIdx.x % bpb) * 256 + tid;
  const size_t qoff = ((size_t)b * N_ + n) * 3;
  const float cx = pts[qoff], cy = pts[qoff + 1], cz = pts[qoff + 2];

  float ls[21], lq[21];
#pragma unroll
  for (int o = 0; o < 21; ++o) { ls[o] = 0.f; lq[o] = 0.f; }
  for (int k = 0; k < K_; ++k) {
    float ex, ey, ez, rx, ry, rz;
    graph_feat(pts, idxbuf, b, n, k, cx, cy, cz, ex, ey, ez, rx, ry, rz);
#pragma unroll
    for (int o = 0; o < 21; ++o) {
      const float f0 = wf[o * 3], f1 = wf[o * 3 + 1], f2 = wf[o * 3 + 2];
      const float px = f0 * ex + f1 * cx + f2 * rx;
      const float py = f0 * ey + f1 * cy + f2 * ry;
      const float pz = f0 * ez + f1 * cz + f2 * rz;
      const float nrm = sqrtf(px * px + py * py + pz * pz) + EPS_;
      ls[o] += nrm; lq[o] += nrm * nrm;
    }
  }
#pragma unroll
  for (int o = 0; o < 21; ++o) {
    atomicAdd(&ssum[2 * o], ls[o]);
    atomicAdd(&ssum[2 * o + 1], lq[o]);
  }
  __syncthreads();
  if (tid < 42) atomicAdd(&stats[tid], ssum[tid]);
}

// writes x1 with channel stride X1CP (pad rows zeroed separately)
__global__ void __launch_bounds__(256)
wpos_apply(const float* __restrict__ pts, const int* __restrict__ idxbuf,
           const float* __restrict__ wf, const float* __restrict__ wd,
           const float* __restrict__ stats, float* __restrict__ x1, int X1CP) {
  const int tid = threadIdx.x;
  const int bpb = N_ / 256;
  const int b = blockIdx.x / bpb;
  const int n = (blockIdx.x % bpb) * 256 + tid;
  const size_t qoff = ((size_t)b * N_ + n) * 3;
  const float cx = pts[qoff], cy = pts[qoff + 1], cz = pts[qoff + 2];
  const float invCnt = 1.f / ((float)B_ * (float)N_ * (float)K_);

  float bn_m[21], bn_r[21];
#pragma unroll
  for (int o = 0; o < 21; ++o) {
    const float mean = stats[2 * o] * invCnt;
    const float var = stats[2 * o + 1] * invCnt - mean * mean;
    bn_m[o] = mean;
    bn_r[o] = 1.f / sqrtf(var + BNEPS_);
  }
  float ax[21], ay[21], az[21];
#pragma unroll
  for (int o = 0; o < 21; ++o) { ax[o] = 0.f; ay[o] = 0.f; az[o] = 0.f; }

  for (int k = 0; k < K_; ++k) {
    float ex, ey, ez, rx, ry, rz;
    graph_feat(pts, idxbuf, b, n, k, cx, cy, cz, ex, ey, ez, rx, ry, rz);
#pragma unroll
    for (int o = 0; o < 21; ++o) {
      const float f0 = wf[o * 3], f1 = wf[o * 3 + 1], f2 = wf[o * 3 + 2];
      float px = f0 * ex + f1 * cx + f2 * rx;
      float py = f0 * ey + f1 * cy + f2 * ry;
      float pz = f0 * ez + f1 * cz + f2 * rz;
      const float nrm = sqrtf(px * px + py * py + pz * pz) + EPS_;
      const float s = ((nrm - bn_m[o]) * bn_r[o]) / nrm;
      px *= s; py *= s; pz *= s;
      const float g0 = wd[o * 3], g1 = wd[o * 3 + 1], g2 = wd[o * 3 + 2];
      const float dx = g0 * ex + g1 * cx + g2 * rx;
      const float dy = g0 * ey + g1 * cy + g2 * ry;
      const float dz = g0 * ez + g1 * cz + g2 * rz;
      const float dot = px * dx + py * dy + pz * dz;
      const float dsq = dx * dx + dy * dy + dz * dz;
      const float t = dot / (dsq + EPS_);
      if (dot >= 0.f) { ax[o] += px; ay[o] += py; az[o] += pz; }
      else            { ax[o] += px - t * dx; ay[o] += py - t * dy; az[o] += pz - t * dz; }
    }
  }
  const float invK = 1.f / (float)K_;
#pragma unroll
  for (int o = 0; o < 21; ++o) {
    const size_t base = ((size_t)(b * X1CP + o) * 3) * N_ + n;
    x1[base] = ax[o] * invK; x1[base + N_] = ay[o] * invK; x1[base + 2 * N_] = az[o] * invK;
  }
}

// ============================================================
// Generic f32 channel GEMM on V_WMMA_F32_16X16X4_F32.
// Wp: zero-padded Mp16 x Kp4 weights. X: padded-channel activations.
// Fully unconditional inner loop: b64 A-load + 2 x b32 B-load + wmma.
// ============================================================
__global__ void __launch_bounds__(128)
wmma_linear(const float* __restrict__ Wp, int Kpad,
            const float* __restrict__ X, int XCP,
            float* __restrict__ Y,
            const float* __restrict__ bias, int biasB) {
  const int lane = threadIdx.x;          // 0..31 (wave32)
  const int half = lane >> 4;            // 0 / 1
  const int lm = lane & 15;
  const int rowBase = blockIdx.y * 16;
  const int colBase = (blockIdx.x * 4 + threadIdx.y) * 16;
  if (colBase >= NC_) return;            // wave-uniform
  const int bz = blockIdx.z;
  const float* Xb = X + (size_t)bz * XCP * NC_;
  float* Yb = Y + (size_t)bz * PCP_ * NC_;
  const int r = rowBase + lm;            // A row (lanes 0-15 / 16-31 both M=0..15)
  const int c = colBase + lm;            // B/C column

  const float* pA = Wp + (size_t)r * Kpad + 2 * half;
  const float* pB = Xb + (size_t)(2 * half) * NC_ + c;
  v8f acc = {};
  const int ksteps = Kpad >> 2;
  for (int ks = 0; ks < ksteps; ++ks) {
    const v2f a = *(const v2f*)pA;                 // W[r][kk], W[r][kk+1]
    v2f bb; bb.x = pB[0]; bb.y = pB[NC_];          // X[kk][c], X[kk+1][c]
    acc = __builtin_amdgcn_wmma_f32_16x16x4_f32(false, a, false, bb,
                                                (short)0, acc, false, false);
    pA += 4; pB += (size_t)4 * NC_;
  }
  const int v = c / N_;                  // vector component of this column
  if (bias) {
    const float* bp = bias + (size_t)bz * biasB;
#pragma unroll
    for (int r8 = 0; r8 < 8; ++r8) {
      const int ro = rowBase + half * 8 + r8;
      Yb[(size_t)ro * NC_ + c] = acc[r8] + bp[ro * 3 + v];
    }
  } else {
#pragma unroll
    for (int r8 = 0; r8 < 8; ++r8) {
      const int ro = rowBase + half * 8 + r8;
      Yb[(size_t)ro * NC_ + c] = acc[r8];
    }
  }
}

// ============================================================
// Global BN stats over ||p|| per channel (count = B*N)
// ============================================================
__global__ void __launch_bounds__(256)
bn_stats(const float* __restrict__ P, float* __restrict__ stats, int C) {
  __shared__ float s1[256], s2[256];
  const int tid = threadIdx.x;
  const int t = blockIdx.x * 256 + tid;            // t in [0, B*C*N)
  const int bo = t / N_;
  const int b = bo / C, o = bo % C;
  const size_t base = ((size_t)(b * PCP_ + o) * 3) * N_ + (t % N_);
  const float px = P[base], py = P[base + N_], pz = P[base + 2 * N_];
  const float nrm = sqrtf(px * px + py * py + pz * pz) + EPS_;
  s1[tid] = nrm; s2[tid] = nrm * nrm;
  __syncthreads();
  for (int st = 128; st > 0; st >>= 1) {
    if (tid < st) { s1[tid] += s1[tid + st]; s2[tid] += s2[tid + st]; }
    __syncthreads();
  }
  if (tid == 0) {
    atomicAdd(&stats[2 * o], s1[0]);
    atomicAdd(&stats[2 * o + 1], s2[0]);
  }
}

// BN + LLR (ns = 0); Y has channel stride YCP
__global__ void __launch_bounds__(256)
bn_llr_apply(const float* __restrict__ P, const float* __restrict__ D,
             float* __restrict__ Y, const float* __restrict__ stats,
             int C, int YCP, float invCnt) {
  const int t = blockIdx.x * 256 + threadIdx.x;
  if (t >= B_ * C * N_) return;
  const int bo = t / N_;
  const int b = bo / C, o = bo % C;
  const int n = t % N_;
  const size_t base = ((size_t)(b * PCP_ + o) * 3) * N_ + n;
  float px = P[base], py = P[base + N_], pz = P[base + 2 * N_];
  const float dx = D[base], dy = D[base + N_], dz = D[base + 2 * N_];
  const float mean = stats[2 * o] * invCnt;
  const float var = stats[2 * o + 1] * invCnt - mean * mean;
  const float nrm = sqrtf(px * px + py * py + pz * pz) + EPS_;
  const float s = ((nrm - mean) / sqrtf(var + BNEPS_)) / nrm;
  px *= s; py *= s; pz *= s;
  const float dot = px * dx + py * dy + pz * dz;
  const float dsq = dx * dx + dy * dy + dz * dz;
  const float tq = dot / (dsq + EPS_);
  const bool pos = dot >= 0.f;
  const size_t yb = ((size_t)(b * YCP + o) * 3) * N_ + n;
  Y[yb]          = pos ? px : px - tq * dx;
  Y[yb + N_]     = pos ? py : py - tq * dy;
  Y[yb + 2 * N_] = pos ? pz : pz - tq * dz;
}

// BN only (w3 stage); Y has channel stride YCP
__global__ void __launch_bounds__(256)
bn_apply(const float* __restrict__ P, float* __restrict__ Y,
         const float* __restrict__ stats, int C, int YCP, float invCnt) {
  const int t = blockIdx.x * 256 + threadIdx.x;
  if (t >= B_ * C * N_) return;
  const int bo = t / N_;
  const int b = bo / C, o = bo % C;
  const int n = t % N_;
  const size_t base = ((size_t)(b * PCP_ + o) * 3) * N_ + n;
  const float px = P[base], py = P[base + N_], pz = P[base + 2 * N_];
  const float mean = stats[2 * o] * invCnt;
  const float var = stats[2 * o + 1] * invCnt - mean * mean;
  const float nrm = sqrtf(px * px + py * py + pz * pz) + EPS_;
  const float s = ((nrm - mean) / sqrtf(var + BNEPS_)) / nrm;
  const size_t yb = ((size_t)(b * YCP + o) * 3) * N_ + n;
  Y[yb] = px * s; Y[yb + N_] = py * s; Y[yb + 2 * N_] = pz * s;
}

// mean over N for each (b, ch, v) -> xmean (ws) and x_mean_out (d_out)
__global__ void __launch_bounds__(256)
mean_over_n(const float* __restrict__ X, int XCP, float* __restrict__ xmean,
            float* __restrict__ out_mean) {
  __shared__ float red[256];
  const int tid = threadIdx.x;
  const int e = blockIdx.x;              // 0 .. B*85*3-1
  const int b = e / (85 * 3), r = e % (85 * 3);
  const int o = r / 3, v = r % 3;
  const size_t base = ((size_t)(b * XCP + o) * 3 + v) * N_;
  float s = 0.f;
  for (int n = tid; n < N_; n += 256) s += X[base + n];
  red[tid] = s;
  __syncthreads();
  for (int st = 128; st > 0; st >>= 1) {
    if (tid < st) red[tid] += red[tid + st];
    __syncthreads();
  }
  if (tid == 0) {
    const float m = red[0] * (1.f / (float)N_);
    xmean[e] = m; out_mean[e] = m;
  }
}

__global__ void center_kernel(const float* __restrict__ xmean, float* __restrict__ outc) {
  const int t = blockIdx.x * blockDim.x + threadIdx.x;
  if (t < B_ * 85)
    outc[t] = (xmean[t * 3] + xmean[t * 3 + 1] + xmean[t * 3 + 2]) * (1.f / 3.f);
}

// bias for ws1: contribution of the broadcast-mean half of the concat
__global__ void fold_mean_bias(const float* __restrict__ ws1f, const float* __restrict__ ws1d,
                               const float* __restrict__ xmean,
                               float* __restrict__ cf, float* __restrict__ cd) {
  const int t = blockIdx.x * blockDim.x + threadIdx.x;
  if (t >= B_ * 85 * 3) return;
  const int b = t / (85 * 3), o = (t / 3) % 85, v = t % 3;
  float sf = 0.f, sd = 0.f;
  for (int c = 0; c < 85; ++c) {
    const float xm = xmean[(b * 85 + c) * 3 + v];
    sf += ws1f[o * 170 + 85 + c] * xm;
    sd += ws1d[o * 170 + 85 + c] * xm;
  }
  cf[t] = sf; cd[t] = sd;
}

// ============================================================
// Final: z = wlin @ z2 (3x3 per point), x_std contraction with
// the 170-ch concat (x4 + broadcast mean), max over N (sortable
// uint atomicMax: LDS first, then global).
// ============================================================
__global__ void __launch_bounds__(256)
final_kernel(const float* __restrict__ z2, int Z2CP, const float* __restrict__ wlin,
             const float* __restrict__ x4, int X4CP, const float* __restrict__ xmean,
             unsigned* __restrict__ inv) {
  __shared__ unsigned smax[510];
  const int tid = threadIdx.x;
  for (int e = tid; e < 510; e += 256) smax[e] = 0x007FFFFFu;  // sortable(-inf)
  __syncthreads();
  const int bpb = N_ / 256;
  const int b = blockIdx.x / bpb;
  const int n = (blockIdx.x % bpb) * 256 + tid;

  float zr[3][3];
#pragma unroll
  for (int i = 0; i < 3; ++i)
#pragma unroll
    for (int j = 0; j < 3; ++j) zr[i][j] = 0.f;
  for (int c = 0; c < 42; ++c) {
    const float w0 = wlin[c], w1 = wlin[42 + c], w2 = wlin[84 + c];
    const size_t base = ((size_t)(b * Z2CP + c) * 3) * N_ + n;
    const float v0 = z2[base], v1 = z2[base + N_], v2 = z2[base + 2 * N_];
    zr[0][0] += w0 * v0; zr[0][1] += w0 * v1; zr[0][2] += w0 * v2;
    zr[1][0] += w1 * v0; zr[1][1] += w1 * v1; zr[1][2] += w1 * v2;
    zr[2][0] += w2 * v0; zr[2][1] += w2 * v1; zr[2][2] += w2 * v2;
  }
  for (int i = 0; i < 170; ++i) {
    float xv0, xv1, xv2;
    if (i < 85) {
      const size_t base = ((size_t)(b * X4CP + i) * 3) * N_ + n;
      xv0 = x4[base]; xv1 = x4[base + N_]; xv2 = x4[base + 2 * N_];
    } else {
      const int c = i - 85;
      xv0 = xmean[(b * 85 + c) * 3];
      xv1 = xmean[(b * 85 + c) * 3 + 1];
      xv2 = xmean[(b * 85 + c) * 3 + 2];
    }
#pragma unroll
    for (int kk = 0; kk < 3; ++kk) {
      const float val = xv0 * zr[kk][0] + xv1 * zr[kk][1] + xv2 * zr[kk][2];
      atomicMax(&smax[i * 3 + kk], sortable(val));
    }
  }
  __syncthreads();
  for (int e = tid; e < 510; e += 256) atomicMax(&inv[b * 510 + e], smax[e]);
}

__global__ void finalize_inv(const unsigned* __restrict__ inv, float* __restrict__ out) {
  const int t = blockIdx.x * blockDim.x + threadIdx.x;
  if (t < B_ * 510) out[t] = unsortable(inv[t]);
}

// ============================================================
// Host launcher
// ============================================================
extern "C" void kernel_launch(void* const* d_in, const int* in_sizes, int n_in,
                              void* d_out, int out_size, void* d_ws, size_t ws_size,
                              hipStream_t stream) {
  (void)in_sizes; (void)n_in; (void)out_size; (void)ws_size;
  const float* points = (const float*)d_in[0];
  const float* wposf  = (const float*)d_in[1];
  const float* wposd  = (const float*)d_in[2];
  const float* w1f    = (const float*)d_in[3];
  const float* w1d    = (const float*)d_in[4];
  const float* w2f    = (const float*)d_in[5];
  const float* w2d    = (const float*)d_in[6];
  const float* w3     = (const float*)d_in[7];
  const float* ws1f   = (const float*)d_in[8];
  const float* ws1d   = (const float*)d_in[9];
  const float* ws2f   = (const float*)d_in[10];
  const float* ws2d   = (const float*)d_in[11];
  const float* wlin   = (const float*)d_in[12];
  float* out = (float*)d_out;

  // padded channel strides
  constexpr int X1CP = 24, X2CP = 24, X3CP = 44, X4CP = 88, Z1CP = 88, Z2CP = 44;

  // workspace layout (4-byte units)
  size_t off = 0;
  auto takeF = [&](size_t n) { float* p = (float*)((char*)d_ws + off * 4); off += n; return p; };
  int*      idxp  = (int*)((char*)d_ws + off * 4);      off += (size_t)B_ * N_ * K_;
  float*    Pbuf  = takeF((size_t)B_ * PCP_ * NC_);
  float*    Dbuf  = takeF((size_t)B_ * PCP_ * NC_);
  float*    x1    = takeF((size_t)B_ * X1CP * NC_);
  float*    x2    = takeF((size_t)B_ * X2CP * NC_);
  float*    x3    = takeF((size_t)B_ * X3CP * NC_);
  float*    x4    = takeF((size_t)B_ * X4CP * NC_);
  float*    z1    = takeF((size_t)B_ * Z1CP * NC_);
  float*    z2    = takeF((size_t)B_ * Z2CP * NC_);
  float*    wpf   = takeF(96 * 88);
  float*    wpd   = takeF(96 * 88);
  float*    stats = takeF(256);
  float*    xmean = takeF(1024);
  float*    cf    = takeF(1024);
  float*    cd    = takeF(1024);
  unsigned* invp  = (unsigned*)((char*)d_ws + off * 4); off += 2048;

  const int nbBN = B_ * N_ / 256;     // 128 blocks of 256 threads covering (b, n)
  const float invBN = 1.f / ((float)B_ * (float)N_);

  // 1) KNN
  knn_kernel<<<nbBN, 256, 0, stream>>>(points, idxp);

  // 2) wpos layer (fused graph feature + 21x3 linears + BN + LLR + mean over K)
  fill_f32<<<1, 64, 0, stream>>>(stats, 42, 0.f);
  wpos_stats<<<nbBN, 256, 0, stream>>>(points, idxp, wposf, stats);
  wpos_apply<<<nbBN, 256, 0, stream>>>(points, idxp, wposf, wposd, stats, x1, X1CP);
  zero_pad<<<(B_ * (X1CP - 21) * 3 * N_ + 255) / 256, 256, 0, stream>>>(x1, 21, X1CP);

  auto llr_layer = [&](const float* Wf, const float* Wd, int ldw, int M, int Kd,
                       const float* Xin, int XCP, float* Yout, int YCP, int YC,
                       const float* biasF, const float* biasD) {
    const int Mp16 = ((M + 15) / 16) * 16;
    const int Kp4 = ((Kd + 3) / 4) * 4;
    pack_w<<<(Mp16 * Kp4 + 255) / 256, 256, 0, stream>>>(Wf, ldw, M, Kd, wpf, Mp16, Kp4);
    pack_w<<<(Mp16 * Kp4 + 255) / 256, 256, 0, stream>>>(Wd, ldw, M, Kd, wpd, Mp16, Kp4);
    dim3 g(NC_ / 16 / 4, Mp16 / 16, B_);
    dim3 bk(32, 4);
    wmma_linear<<<g, bk, 0, stream>>>(wpf, Kp4, Xin, XCP, Pbuf, biasF, 85 * 3);
    wmma_linear<<<g, bk, 0, stream>>>(wpd, Kp4, Xin, XCP, Dbuf, biasD, 85 * 3);
    fill_f32<<<1, 256, 0, stream>>>(stats, 2 * M, 0.f);
    bn_stats<<<B_ * M * N_ / 256, 256, 0, stream>>>(Pbuf, stats, M);
    bn_llr_apply<<<B_ * M * N_ / 256, 256, 0, stream>>>(Pbuf, Dbuf, Yout, stats, M, YCP, invBN);
    if (YCP > YC)
      zero_pad<<<(B_ * (YCP - YC) * 3 * N_ + 255) / 256, 256, 0, stream>>>(Yout, YC, YCP);
  };

  // 3) w1: 21 -> 21 (BN + LLR)
  llr_layer(w1f, w1d, 21, 21, 21, x1, X1CP, x2, X2CP, 21, nullptr, nullptr);
  // 4) w2[:, :21]: 21 -> 42 (BN + LLR)
  llr_layer(w2f, w2d, 42, 42, 21, x2, X2CP, x3, X3CP, 42, nullptr, nullptr);

  // 5) w3: 42 -> 85, BN only
  {
    const int Mp16 = 96, Kp4 = 44;
    pack_w<<<(Mp16 * Kp4 + 255) / 256, 256, 0, stream>>>(w3, 42, 85, 42, wpf, Mp16, Kp4);
    dim3 g(NC_ / 16 / 4, Mp16 / 16, B_);
    dim3 bk(32, 4);
    wmma_linear<<<g, bk, 0, stream>>>(wpf, Kp4, x3, X3CP, Pbuf, nullptr, 0);
    fill_f32<<<1, 256, 0, stream>>>(stats, 170, 0.f);
    bn_stats<<<B_ * 85 * N_ / 256, 256, 0, stream>>>(Pbuf, stats, 85);
    bn_apply<<<B_ * 85 * N_ / 256, 256, 0, stream>>>(Pbuf, x4, stats, 85, X4CP, invBN);
    zero_pad<<<(B_ * (X4CP - 85) * 3 * N_ + 255) / 256, 256, 0, stream>>>(x4, 85, X4CP);
  }

  // 6) x_mean_out + center_loc (direct outputs) and concat folded into bias
  mean_over_n<<<B_ * 85 * 3, 256, 0, stream>>>(x4, X4CP, xmean, out + B_ * 510);
  center_kernel<<<2, 256, 0, stream>>>(xmean, out + B_ * 510 + B_ * 85 * 3);
  fold_mean_bias<<<4, 256, 0, stream>>>(ws1f, ws1d, xmean, cf, cd);

  // 7) ws1: (85 | mean-broadcast 85) -> 85 (BN + LLR); concat half folded to bias
  llr_layer(ws1f, ws1d, 170, 85, 85, x4, X4CP, z1, Z1CP, 85, cf, cd);
  // 8) ws2: 85 -> 42 (BN + LLR)
  llr_layer(ws2f, ws2d, 85, 42, 85, z1, Z1CP, z2, Z2CP, 42, nullptr, nullptr);

  // 9) final: wlin + frame contraction + max over N
  fill_u32<<<(B_ * 510 + 255) / 256, 256, 0, stream>>>(invp, B_ * 510, 0x007FFFFFu);
  final_kernel<<<nbBN, 256, 0, stream>>>(z2, Z2CP, wlin, x4, X4CP, xmean, invp);
  finalize_inv<<<(B_ * 510 + 255) / 256, 256, 0, stream>>>(invp, out);
}